// MEMFNet_26207890440587
// MI455X (gfx1250) — compile-verified
//
#include <hip/hip_runtime.h>
#include <math.h>

typedef __attribute__((ext_vector_type(2))) float v2f;
typedef __attribute__((ext_vector_type(8))) float v8f;

#define N_NODES 65536
#define EPC 8
#define N_CRY (N_NODES / EPC)
#define EMB 200
#define FEA 64
#define HID 256
#define NHEAD 3
#define EPSV 1e-10f

// padded LDS row strides (floats); all even so v2f loads stay 8B aligned
#define SH 260   // hidden matrix rows (256 cols + pad)  : 260 % 64 == 4  -> conflict-free
#define SX 68    // x rows (64 cols + pad)               : 68  % 64 == 4
#define SM 66    // msg rows (64 cols + pad)

__device__ __forceinline__ float silu_f(float x) { return x / (1.0f + expf(-x)); }
__device__ __forceinline__ v2f ldv2(const float* p) { return *(const v2f*)p; }

#define WMMA_F32(A, B, C) \
  __builtin_amdgcn_wmma_f32_16x16x4_f32(false, (A), false, (B), (short)0, (C), false, false)

// 16 K-steps (K=64 slice), 4 independent N-column-group accumulators, software pipelined.
__device__ __forceinline__ void mm_pass16(const float* __restrict__ arow,
                                          const float* __restrict__ w0,
                                          const float* __restrict__ w1,
                                          const float* __restrict__ w2,
                                          const float* __restrict__ w3, int lhi2,
                                          v8f& c0, v8f& c1, v8f& c2, v8f& c3) {
  v2f a  = ldv2(arow + lhi2);
  v2f b0 = ldv2(w0 + lhi2), b1 = ldv2(w1 + lhi2), b2 = ldv2(w2 + lhi2), b3 = ldv2(w3 + lhi2);
#pragma unroll 5
  for (int kk = 0; kk < 15; ++kk) {
    const int o = (kk + 1) * 4 + lhi2;
    v2f an = ldv2(arow + o);
    v2f n0 = ldv2(w0 + o), n1 = ldv2(w1 + o), n2 = ldv2(w2 + o), n3 = ldv2(w3 + o);
    c0 = WMMA_F32(a, b0, c0);
    c1 = WMMA_F32(a, b1, c1);
    c2 = WMMA_F32(a, b2, c2);
    c3 = WMMA_F32(a, b3, c3);
    a = an; b0 = n0; b1 = n1; b2 = n2; b3 = n3;
  }
  c0 = WMMA_F32(a, b0, c0);
  c1 = WMMA_F32(a, b1, c1);
  c2 = WMMA_F32(a, b2, c2);
  c3 = WMMA_F32(a, b3, c3);
}

// 64 K-steps (K=256), single accumulator, software pipelined.
__device__ __forceinline__ void mm_pass_g2(const float* __restrict__ hrow,
                                           const float* __restrict__ brow, int lhi2, v8f& c) {
  v2f a = ldv2(hrow + lhi2);
  v2f b = ldv2(brow + lhi2);
#pragma unroll 7
  for (int kk = 0; kk < 63; ++kk) {
    const int o = (kk + 1) * 4 + lhi2;
    v2f an = ldv2(hrow + o);
    v2f bn = ldv2(brow + o);
    c = WMMA_F32(a, b, c);
    a = an; b = bn;
  }
  c = WMMA_F32(a, b, c);
}

// ---------------- embedding: x[:, :63] = fea @ W^T + b ; x[:,63] = elem_weights ----------
__global__ void embed_kernel(const float* __restrict__ fea,
                             const float* __restrict__ W,   // (63,200)
                             const float* __restrict__ b,   // (63,)
                             const float* __restrict__ ew,  // (N,)
                             float* __restrict__ x) {       // (N,64)
  __shared__ float s_fea[4 * EMB];
  int t = threadIdx.x;
  int n0 = blockIdx.x * 4;
  for (int i = t; i < 4 * EMB; i += 256) s_fea[i] = fea[(size_t)n0 * EMB + i];
  __syncthreads();
  int ln = t >> 6, f = t & 63;
  int n = n0 + ln;
  float out;
  if (f == 63) {
    out = ew[n];
  } else {
    float s = 0.f;
    const float* wr = W + f * EMB;
    const float* fr = s_fea + ln * EMB;
    for (int k = 0; k < EMB; ++k) s += fr[k] * wr[k];
    out = s + b[f];
  }
  x[(size_t)n * 64 + f] = out;
}

// ---------------- fused attention layer (message or crystal pooling) ---------------------
// CRYSTAL=false: 1 crystal/block, 64 edge-rows, pair K=128, segments = 8 nodes, residual.
// CRYSTAL=true : 8 crystals/block, 64 node-rows, K=64, segments = 8 crystals.
template <int K, bool CRYSTAL>
__global__ void __launch_bounds__(512)
attn_kernel(const float* __restrict__ x_in,
            const float* __restrict__ ew,
            const float* __restrict__ gW1, const float* __restrict__ gb1,
            const float* __restrict__ gW2, const float* __restrict__ gb2,
            const float* __restrict__ mW1, const float* __restrict__ mb1,
            const float* __restrict__ mW2, const float* __restrict__ mb2,
            float* __restrict__ x_out) {
  __shared__ float s_x[(CRYSTAL ? 64 : 8) * SX];
  __shared__ float s_h[64 * SH];    // full SiLU(hidden): 64 rows x 256 (+pad)
  __shared__ float s_msg[64 * SM];  // message-net output
  __shared__ float s_gate[64];
  __shared__ float s_red[512];
  __shared__ float s_w[64];

  const int t = threadIdx.x;
  const int lane = t & 31;
  const int wv = t >> 5;            // 16 wave32s
  const int lh = lane & 15;
  const int lhi2 = (lane >> 4) * 2; // 0 or 2: K sub-offset for high half-lanes
  const int Mt = wv & 3;            // 4 M-tiles of 16 rows
  const int Nt = wv >> 2;           // 4 N-tiles within each 64-col group
  const int r0 = Mt * 16;
  const int r = r0 + lh;            // this lane's matrix row

  // ---- stage x tile into (padded) LDS with async global->LDS b128 copies ----
  {
    const int nbase = blockIdx.x * (CRYSTAL ? 64 : EPC);
    const int nrow16 = CRYSTAL ? 1024 : 128;  // rows*16 transfers of 16B each
    const unsigned sxb = (unsigned)(size_t)(__builtin_addressof(s_x[0]));
    for (int i = t; i < nrow16; i += 512) {
      const int row = i >> 4, c4 = i & 15;
      unsigned ldsb = sxb + (unsigned)(row * SX + c4 * 4) * 4u;
      unsigned long long ga =
          (unsigned long long)(size_t)(x_in + (size_t)(nbase + row) * 64 + c4 * 4);
      asm volatile("global_load_async_to_lds_b128 %0, %1, off"
                   :: "v"(ldsb), "v"(ga) : "memory");
    }
    if (t < (CRYSTAL ? 64 : 8)) s_w[t] = ew[nbase + t];
    asm volatile("s_wait_asynccnt 0x0" ::: "memory");
  }
  float hacc = 0.f;
  __syncthreads();

  for (int h = 0; h < NHEAD; ++h) {
    const float* W1  = gW1 + (size_t)h * HID * K;
    const float* b1  = gb1 + h * HID;
    const float* w2  = gW2 + h * HID;
    const float  bb2 = gb2[h];
    const float* M1  = mW1 + (size_t)h * HID * K;
    const float* B1m = mb1 + h * HID;
    const float* M2  = mW2 + (size_t)h * FEA * HID;
    const float* B2m = mb2 + h * FEA;
    const int nb = Nt * 16 + lh;  // base hidden unit for this wave/lane

    // ================= gate net: SiLU((64xK) @ W1^T + b1) . w2 + b2 =================
    {
      v8f c0 = {}, c1 = {}, c2 = {}, c3 = {};
      const float* w0p = W1 + (size_t)(nb)*K;
      const float* w1p = W1 + (size_t)(nb + 64) * K;
      const float* w2p = W1 + (size_t)(nb + 128) * K;
      const float* w3p = W1 + (size_t)(nb + 192) * K;
      if (CRYSTAL) {
        mm_pass16(s_x + r * SX, w0p, w1p, w2p, w3p, lhi2, c0, c1, c2, c3);
      } else {
        mm_pass16(s_x + (r >> 3) * SX, w0p, w1p, w2p, w3p, lhi2, c0, c1, c2, c3);
        mm_pass16(s_x + (r & 7) * SX, w0p + 64, w1p + 64, w2p + 64, w3p + 64, lhi2,
                  c0, c1, c2, c3);
      }
      v8f cc[4] = {c0, c1, c2, c3};
#pragma unroll
      for (int q = 0; q < 4; ++q) {
        const int nq = nb + q * 64;
        const float bias = b1[nq];
#pragma unroll
        for (int v = 0; v < 8; ++v) {
          int row = r0 + v + (lhi2 << 2);  // +8 for high half-lanes
          s_h[row * SH + nq] = silu_f(cc[q][v] + bias);
        }
      }
    }
    __syncthreads();
    {  // dot with w2 (HID->1): 8 partial lanes per row
      const int e = t >> 3, sub = t & 7;
      const float* w2c = w2 + sub * 32;
      const float* hrow = s_h + e * SH + sub * 32;
      float pg = 0.f;
#pragma unroll
      for (int k = 0; k < 32; ++k) pg += hrow[k] * w2c[k];
      s_red[t] = pg;
    }
    __syncthreads();
    if (t < 64) {
      float g = bb2;
#pragma unroll
      for (int s = 0; s < 8; ++s) g += s_red[t * 8 + s];
      s_gate[t] = g;
    }
    __syncthreads();

    // ================= message net: SiLU((64xK)@M1^T + b1) @ M2^T + b2 =================
    {
      v8f c0 = {}, c1 = {}, c2 = {}, c3 = {};
      const float* w0p = M1 + (size_t)(nb)*K;
      const float* w1p = M1 + (size_t)(nb + 64) * K;
      const float* w2p = M1 + (size_t)(nb + 128) * K;
      const float* w3p = M1 + (size_t)(nb + 192) * K;
      if (CRYSTAL) {
        mm_pass16(s_x + r * SX, w0p, w1p, w2p, w3p, lhi2, c0, c1, c2, c3);
      } else {
        mm_pass16(s_x + (r >> 3) * SX, w0p, w1p, w2p, w3p, lhi2, c0, c1, c2, c3);
        mm_pass16(s_x + (r & 7) * SX, w0p + 64, w1p + 64, w2p + 64, w3p + 64, lhi2,
                  c0, c1, c2, c3);
      }
      v8f cc[4] = {c0, c1, c2, c3};
#pragma unroll
      for (int q = 0; q < 4; ++q) {
        const int nq = nb + q * 64;
        const float bias = B1m[nq];
#pragma unroll
        for (int v = 0; v < 8; ++v) {
          int row = r0 + v + (lhi2 << 2);
          s_h[row * SH + nq] = silu_f(cc[q][v] + bias);
        }
      }
    }
    __syncthreads();
    {  // second GEMM: (64x256) @ M2^T -> (64x64)
      v8f mc = {};
      mm_pass_g2(s_h + r * SH, M2 + (size_t)nb * HID, lhi2, mc);
      const float bias = B2m[nb];
#pragma unroll
      for (int v = 0; v < 8; ++v) {
        int row = r0 + v + (lhi2 << 2);
        s_msg[row * SM + nb] = mc[v] + bias;
      }
    }
    __syncthreads();

    // ================= segment softmax pooling (8 contiguous rows / segment) ============
    {
      const int seg = t >> 6, f = t & 63;
      float gmax = -3.0e38f;
#pragma unroll
      for (int j = 0; j < 8; ++j) gmax = fmaxf(gmax, s_gate[seg * 8 + j]);
      float den = 0.f, num = 0.f;
#pragma unroll
      for (int j = 0; j < 8; ++j) {
        float wj = CRYSTAL ? s_w[seg * 8 + j] : s_w[j];
        float e = wj * expf(s_gate[seg * 8 + j] - gmax);
        den += e;
        num += e * s_msg[(seg * 8 + j) * SM + f];
      }
      hacc += num / (den + EPSV);
    }
    __syncthreads();
  }

  {
    const int seg = t >> 6, f = t & 63;
    if (CRYSTAL) {
      int c = blockIdx.x * 8 + seg;
      x_out[(size_t)c * 64 + f] = hacc * (1.0f / 3.0f);
    } else {
      int node = blockIdx.x * EPC + seg;
      x_out[(size_t)node * 64 + f] = hacc * (1.0f / 3.0f) + s_x[seg * SX + f];
    }
  }
}

extern "C" void kernel_launch(void* const* d_in, const int* in_sizes, int n_in,
                              void* d_out, int out_size, void* d_ws, size_t ws_size,
                              hipStream_t stream) {
  const float* ew    = (const float*)d_in[0];
  const float* fea   = (const float*)d_in[1];
  const float* W_emb = (const float*)d_in[2];
  const float* b_emb = (const float*)d_in[3];
  const float* gW1   = (const float*)d_in[4];
  const float* gb1   = (const float*)d_in[5];
  const float* gW2   = (const float*)d_in[6];
  const float* gb2   = (const float*)d_in[7];
  const float* mW1   = (const float*)d_in[8];
  const float* mb1   = (const float*)d_in[9];
  const float* mW2   = (const float*)d_in[10];
  const float* mb2   = (const float*)d_in[11];
  const float* cgW1  = (const float*)d_in[12];
  const float* cgb1  = (const float*)d_in[13];
  const float* cgW2  = (const float*)d_in[14];
  const float* cgb2  = (const float*)d_in[15];
  const float* cmW1  = (const float*)d_in[16];
  const float* cmb1  = (const float*)d_in[17];
  const float* cmW2  = (const float*)d_in[18];
  const float* cmb2  = (const float*)d_in[19];

  float* x0 = (float*)d_ws;                       // 16 MB
  float* x1 = x0 + (size_t)N_NODES * 64;          // 16 MB
  float* cry_out = (float*)d_out;                 // (C,64)
  float* x_final = cry_out + (size_t)N_CRY * 64;  // (N,64), second tuple output

  embed_kernel<<<N_NODES / 4, 256, 0, stream>>>(fea, W_emb, b_emb, ew, x0);

  float* xs[4] = {x0, x1, x0, x_final};
  for (int g = 0; g < 3; ++g) {
    attn_kernel<128, false><<<N_CRY, 512, 0, stream>>>(
        xs[g], ew,
        gW1 + (size_t)g * 3 * HID * 128, gb1 + g * 3 * HID,
        gW2 + g * 3 * HID,               gb2 + g * 3,
        mW1 + (size_t)g * 3 * HID * 128, mb1 + g * 3 * HID,
        mW2 + (size_t)g * 3 * FEA * HID, mb2 + g * 3 * FEA,
        xs[g + 1]);
  }

  attn_kernel<64, true><<<N_CRY / 8, 512, 0, stream>>>(
      x_final, ew, cgW1, cgb1, cgW2, cgb2, cmW1, cmb1, cmW2, cmb2, cry_out);
}